// DynamicRouting_9466107920833
// MI455X (gfx1250) — compile-verified
//
#include <hip/hip_runtime.h>
#include <cstdint>
#include <cstddef>

// Problem shape (fixed by the reference)
#define B_SZ   128
#define I_CAPS 2048
#define O_CAPS 32
#define POSE   4
#define ITERS  3
#define EPS    1e-7f

// Kernel tiling: deep async pipeline to cover HBM latency-BW product.
#define THREADS 512
#define WAVES   (THREADS / 32)     // 16 waves, wave32
#define TILE    64                 // in-caps staged per tile
#define IPW     (TILE / WAVES)     // in-caps per wave per tile = 4
#define NT      (I_CAPS / TILE)    // 32 tiles
#define NBUF    4                  // ring buffer
#define DIST    3                  // prefetch distance (tiles ahead)

typedef float v2f __attribute__((ext_vector_type(2)));
typedef float v8f __attribute__((ext_vector_type(8)));

// Generic pointer -> 32-bit LDS byte offset (addrspacecast to AS(3), then ptrtoint).
__device__ __forceinline__ unsigned lds_off(const void* p) {
  return (unsigned)(unsigned long long)(__attribute__((address_space(3))) const void*)p;
}

__launch_bounds__(THREADS, 1)
__global__ void caps_route_kernel(const float* __restrict__ votes,
                                  float* __restrict__ out)
{
  // 4 * 64 * 32 * 4 * 4B = 128 KB staging + 8 KB partials + 1 KB state (~137 KB,
  // within CDNA5's 320 KB-per-workgroup LDS).
  __shared__ __align__(16) float sbuf[NBUF][TILE][O_CAPS][POSE];
  __shared__ __align__(16) float sred[O_CAPS * POSE][WAVES];
  __shared__ __align__(16) float vacc[O_CAPS][POSE];
  __shared__ __align__(16) float stot[O_CAPS * POSE];

  const int b    = blockIdx.x;          // batch element
  const int tid  = threadIdx.x;
  const int lane = tid & 31;            // lane == output capsule o
  const int w    = tid >> 5;            // wave id

  const unsigned long long gbase =
      (unsigned long long)(uintptr_t)votes +
      (unsigned long long)b * (unsigned long long)(I_CAPS * O_CAPS * POSE) * 4ull;

  if (w == 0) {
    vacc[lane][0] = 0.f; vacc[lane][1] = 0.f;
    vacc[lane][2] = 0.f; vacc[lane][3] = 0.f;
  }
  __syncthreads();

  for (int iter = 0; iter < ITERS; ++iter) {
    // V_acc for this lane's capsule (b_ij == dot(votes, V_acc) at this iteration)
    const float va0 = vacc[lane][0], va1 = vacc[lane][1];
    const float va2 = vacc[lane][2], va3 = vacc[lane][3];
    float sp0 = 0.f, sp1 = 0.f, sp2 = 0.f, sp3 = 0.f;

    // ---- prologue: async-stage tiles 0..DIST-1 into LDS (this wave's rows only)
    for (int pt = 0; pt < DIST; ++pt) {
      #pragma unroll
      for (int j = 0; j < IPW; ++j) {
        const int li = w * IPW + j;
        const int ig = pt * TILE + li;
        const unsigned laddr = lds_off(&sbuf[pt][li][lane][0]);
        const unsigned long long ga =
            gbase + (unsigned long long)((ig * O_CAPS + lane) * 16);
        asm volatile("global_load_async_to_lds_b128 %0, %1, off"
                     :: "v"(laddr), "v"(ga) : "memory");
      }
    }

    for (int t = 0; t < NT; ++t) {
      // ---- prefetch tile t+DIST (buffer reused from tile t-1, already consumed)
      if (t + DIST < NT) {
        const int tn = t + DIST;
        const int bs = tn % NBUF;
        #pragma unroll
        for (int j = 0; j < IPW; ++j) {
          const int li = w * IPW + j;
          const int ig = tn * TILE + li;
          const unsigned laddr = lds_off(&sbuf[bs][li][lane][0]);
          const unsigned long long ga =
              gbase + (unsigned long long)((ig * O_CAPS + lane) * 16);
          asm volatile("global_load_async_to_lds_b128 %0, %1, off"
                       :: "v"(laddr), "v"(ga) : "memory");
        }
      }
      // ---- wait until tile t is resident (async loads complete in order per wave);
      // threshold = IPW * (tiles still allowed outstanding)
      const int ahead = (NT - 1 - t) < DIST ? (NT - 1 - t) : DIST;
      if (ahead == 3)      asm volatile("s_wait_asynccnt 12" ::: "memory");
      else if (ahead == 2) asm volatile("s_wait_asynccnt 8"  ::: "memory");
      else if (ahead == 1) asm volatile("s_wait_asynccnt 4"  ::: "memory");
      else                 asm volatile("s_wait_asynccnt 0"  ::: "memory");

      // ---- compute tile t (wave-private rows; no inter-wave barrier needed)
      const int bs = t % NBUF;
      #pragma unroll
      for (int j = 0; j < IPW; ++j) {
        const int li = w * IPW + j;
        const float4 vv = *(const float4*)&sbuf[bs][li][lane][0];
        // b_io = dot4(votes[i,o,:], V_acc[o,:])
        float bb = vv.x * va0 + vv.y * va1 + vv.z * va2 + vv.w * va3;
        // softmax over O == full wave32 reduction (lane == o)
        float m = bb;
        #pragma unroll
        for (int off = 16; off > 0; off >>= 1)
          m = fmaxf(m, __shfl_xor(m, off, 32));
        const float e = __expf(bb - m);
        float s = e;
        #pragma unroll
        for (int off = 16; off > 0; off >>= 1)
          s += __shfl_xor(s, off, 32);
        const float c = e / s;
        // s_j partial accumulation
        sp0 += c * vv.x; sp1 += c * vv.y; sp2 += c * vv.z; sp3 += c * vv.w;
      }
    }

    // ---- cross-wave reduction of s_j partials: layout [col = o*4+p][wave]
    sred[lane * 4 + 0][w] = sp0;
    sred[lane * 4 + 1][w] = sp1;
    sred[lane * 4 + 2][w] = sp2;
    sred[lane * 4 + 3][w] = sp3;
    __syncthreads();

    if (w == 0) {
#if __has_builtin(__builtin_amdgcn_wmma_f32_16x16x4_f32)
      // Exact ones-matrix reduction on the matrix core:
      // D[r,c] = sum_K 1.0 * B[K,c]  -> 16-wave sum per column, chained K=4 x 4.
      const v2f ones = {1.f, 1.f};
      const int sub  = (lane >> 4) << 1;           // lanes 16..31 carry K=2,3 rows
      #pragma unroll
      for (int g = 0; g < 8; ++g) {                // 8 groups of 16 columns
        v8f d = {0.f, 0.f, 0.f, 0.f, 0.f, 0.f, 0.f, 0.f};
        const int col = g * 16 + (lane & 15);
        #pragma unroll
        for (int tt = 0; tt < 4; ++tt) {
          const int kw = tt * 4 + sub;             // wave indices this lane supplies
          const float2 bv = *(const float2*)&sred[col][kw];
          v2f bvec = {bv.x, bv.y};
          d = __builtin_amdgcn_wmma_f32_16x16x4_f32(false, ones, false, bvec,
                                                    (short)0, d, false, false);
        }
        if (lane < 16) stot[g * 16 + lane] = d[0];
      }
#else
      #pragma unroll
      for (int g = 0; g < 4; ++g) {
        const int col = g * 32 + lane;
        float acc = 0.f;
        #pragma unroll
        for (int ww = 0; ww < WAVES; ++ww) acc += sred[col][ww];
        stot[col] = acc;
      }
#endif
      // ---- squash + V_acc update + (final iter) outputs, all per-lane (per-o)
      const float4 st = *(const float4*)&stot[lane * 4];
      const float nsq = st.x * st.x + st.y * st.y + st.z * st.z + st.w * st.w + EPS;
      const float k   = sqrtf(nsq) / (1.f + nsq);  // == scale / s_norm
      const float v0 = st.x * k, v1 = st.y * k, v2 = st.z * k, v3 = st.w * k;
      vacc[lane][0] = va0 + v0;
      vacc[lane][1] = va1 + v1;
      vacc[lane][2] = va2 + v2;
      vacc[lane][3] = va3 + v3;
      if (iter == ITERS - 1) {
        float4 pv; pv.x = v0; pv.y = v1; pv.z = v2; pv.w = v3;
        *(float4*)&out[(size_t)b * (O_CAPS * POSE) + lane * POSE] = pv;
        out[(size_t)B_SZ * O_CAPS * POSE + (size_t)b * O_CAPS + lane] =
            sqrtf(v0 * v0 + v1 * v1 + v2 * v2 + v3 * v3 + EPS);
      }
    }
    __syncthreads();
  }
}

extern "C" void kernel_launch(void* const* d_in, const int* in_sizes, int n_in,
                              void* d_out, int out_size, void* d_ws, size_t ws_size,
                              hipStream_t stream) {
  (void)in_sizes; (void)n_in; (void)out_size; (void)d_ws; (void)ws_size;
  const float* votes = (const float*)d_in[0];   // [128, 2048, 32, 4] f32
  // d_in[1] = activations_in: unused by the reference computation.
  float* out = (float*)d_out;                   // 16384 poses + 4096 activations
  caps_route_kernel<<<B_SZ, THREADS, 0, stream>>>(votes, out);
}